// AlternatingFlow_38079180046712
// MI455X (gfx1250) — compile-verified
//
#include <hip/hip_runtime.h>
#include <math.h>

// ---------------------------------------------------------------------------
// AlternatingFlow fused kernel for gfx1250 (MI455X)
//   L=16 alternating planar/radial layers, B=131072, D=256, fp32.
//   z trajectory is per-row independent -> fully fused in registers.
//   Planar z@w runs through V_WMMA_F32_16X16X4_F32 (broadcast-B trick),
//   split into 4 independent accumulation chains for XDL ILP.
// ---------------------------------------------------------------------------

#define LFLOW  16
#define NPAIR  8          // 8 planar + 8 radial, strictly alternating
#define BATCH  131072
#define DIM    256
#define EPSF   1e-7f

#define NBLK_MAIN (BATCH / 128)   // 1024 blocks, 8 waves * 16 rows each
#define NBLK_LOSS (BATCH / 256)   // 512 blocks

typedef float v2f __attribute__((ext_vector_type(2)));
typedef float v8f __attribute__((ext_vector_type(8)));

// workspace layout (float offsets)
#define OFF_SCORR    ((size_t)0)                      // 8*256 corrected s
#define OFF_WSDOT    ((size_t)2048)                   // 8 dot(w, s_corr)
#define OFF_BETA     ((size_t)2056)                   // 8 corrected beta
#define OFF_R        ((size_t)2064)                   // 8 Frobenius norms R
#define OFF_R2P      ((size_t)2072)                   // NBLK_MAIN*8 r^2 partials
#define OFF_LOSSROW  ((size_t)(2072 + NBLK_MAIN * 8)) // BATCH planar loss/row
#define OFF_H        (OFF_LOSSROW + (size_t)BATCH)    // 8*BATCH stored h
#define OFF_LOSSPART (OFF_H + (size_t)8 * BATCH)      // NBLK_LOSS partial sums

#if defined(__has_builtin)
#  if __has_builtin(__builtin_amdgcn_wmma_f32_16x16x4_f32)
#    define USE_WMMA_F32X4 1
#  endif
#endif
#ifndef USE_WMMA_F32X4
#  define USE_WMMA_F32X4 0
#endif

__device__ __forceinline__ float bred(float v, float* red) {
    const int t = threadIdx.x;
    red[t] = v;
    __syncthreads();
#pragma unroll
    for (int s = 128; s >= 1; s >>= 1) {
        if (t < s) red[t] += red[t + s];
        __syncthreads();
    }
    float r = red[0];
    __syncthreads();
    return r;
}

// ---- prep: corrected s vectors, dot(w,s'), corrected beta (1 block) --------
__global__ __launch_bounds__(256) void flow_prep(const float* __restrict__ pw,
                                                 const float* __restrict__ ps,
                                                 const float* __restrict__ ra,
                                                 const float* __restrict__ rb,
                                                 float* __restrict__ ws) {
    __shared__ float red[256];
    const int c = threadIdx.x;
    for (int p = 0; p < NPAIR; ++p) {
        const float* w = pw + (2 * p) * DIM;   // planar layers use even rows
        const float* s = ps + (2 * p) * DIM;
        float wv = w[c], sv = s[c];
        float margin = bred(sv * wv, red);
        float wn2    = bred(wv * wv, red);
        float comp = -1.f + log1pf(expf(margin)) - margin;
        float scv  = (margin < -1.f) ? (sv + comp * wv / sqrtf(wn2)) : sv;
        ws[OFF_SCORR + p * DIM + c] = scv;
        float wsd = bred(wv * scv, red);
        if (c == 0) ws[OFF_WSDOT + p] = wsd;
    }
    if (c < NPAIR) {
        float a = ra[2 * c + 1], bt = rb[2 * c + 1];  // radial uses odd rows
        ws[OFF_BETA + c] = (bt < -a) ? (-a + log1pf(expf(bt))) : bt;
    }
}

// ---- main: all 16 layers fused, z resident in registers --------------------
__global__ __launch_bounds__(256) void flow_main(const float* __restrict__ zin,
                                                 const float* __restrict__ pw,
                                                 const float* __restrict__ pb,
                                                 const float* __restrict__ rz0,
                                                 const float* __restrict__ ra,
                                                 float* __restrict__ zout,
                                                 float* __restrict__ ws) {
    __shared__ v2f   s_wswz[NPAIR * 2 * 64];   // w in WMMA-B broadcast layout
    __shared__ float s_scorr[NPAIR * DIM];
    __shared__ float s_z0[NPAIR * DIM];
    __shared__ float s_b[NPAIR], s_wsdot[NPAIR], s_beta[NPAIR], s_a[NPAIR];
    __shared__ float s_wr2[8][NPAIR];          // [wave][radial layer]

    const int tid = threadIdx.x;
    // cooperative LDS fill
    for (int idx = tid; idx < NPAIR * 2 * 64; idx += 256) {
        int p = idx >> 7, rem = idx & 127, hf = rem >> 6, t = rem & 63;
        const float* w = pw + (2 * p) * DIM;
        v2f v;
        v.x = w[4 * t + hf];       // B row (hf)   : k = hf
        v.y = w[4 * t + 2 + hf];   // B row (hf+2) : k = hf+2
        s_wswz[idx] = v;
    }
    for (int idx = tid; idx < NPAIR * DIM; idx += 256)
        s_scorr[idx] = ws[OFF_SCORR + idx];
    for (int idx = tid; idx < NPAIR * DIM; idx += 256) {
        int q = idx >> 8, c = idx & 255;
        s_z0[idx] = rz0[(2 * q + 1) * DIM + c];
    }
    if (tid < NPAIR) {
        s_b[tid]     = pb[2 * tid];
        s_wsdot[tid] = ws[OFF_WSDOT + tid];
        s_beta[tid]  = ws[OFF_BETA + tid];
        s_a[tid]     = ra[2 * tid + 1];
    }
    __syncthreads();

    const int wave  = tid >> 5, lane = tid & 31;
    const int half  = lane >> 4, myrow = lane & 15;
    const int grow  = blockIdx.x * 128 + wave * 16 + myrow;  // global row

    // z tile in WMMA f32 A-layout: lane<16 holds cols {4t,4t+1} of row myrow,
    // lane>=16 holds cols {4t+2,4t+3}. 64 float2 chunks = 128 VGPRs.
    v2f zc[64];
    const v2f* zin2 = reinterpret_cast<const v2f*>(zin) + (size_t)grow * (DIM / 2) + half;
#pragma unroll
    for (int t = 0; t < 64; ++t) zc[t] = zin2[t * 2];

    float loss = 0.f;
    const v2f* scorr2 = reinterpret_cast<const v2f*>(s_scorr);
    const v2f* z02    = reinterpret_cast<const v2f*>(s_z0);

    for (int i = 0; i < NPAIR; ++i) {
        // ======================= planar layer (2i) =======================
        float pre;
#if USE_WMMA_F32X4
        // C[m,n] accumulates dot(z[m,:], w) with w broadcast across B columns.
        // 4 independent accumulation chains -> 4x ILP on the matrix pipe.
        v8f c0 = {}, c1 = {}, c2 = {}, c3 = {};
        const v2f* wb = &s_wswz[(i * 2 + half) * 64];
#pragma unroll
        for (int t = 0; t < 16; ++t) {
            c0 = __builtin_amdgcn_wmma_f32_16x16x4_f32(
                false, zc[4 * t + 0], false, wb[4 * t + 0], (short)0, c0, false, false);
            c1 = __builtin_amdgcn_wmma_f32_16x16x4_f32(
                false, zc[4 * t + 1], false, wb[4 * t + 1], (short)0, c1, false, false);
            c2 = __builtin_amdgcn_wmma_f32_16x16x4_f32(
                false, zc[4 * t + 2], false, wb[4 * t + 2], (short)0, c2, false, false);
            c3 = __builtin_amdgcn_wmma_f32_16x16x4_f32(
                false, zc[4 * t + 3], false, wb[4 * t + 3], (short)0, c3, false, false);
        }
        v8f cacc = (c0 + c1) + (c2 + c3);
        // extract column 0 of C: m<8 -> vgpr m lane 0; m>=8 -> vgpr m-8 lane 16
        pre = 0.f;
#pragma unroll
        for (int r = 0; r < 8; ++r) {
            float v0 = __shfl(cacc[r], 0, 32);
            float v1 = __shfl(cacc[r], 16, 32);
            if (myrow == r)     pre = v0;
            if (myrow == r + 8) pre = v1;
        }
#else
        float part = 0.f;
#pragma unroll
        for (int t = 0; t < 64; ++t) {
            v2f a0 = s_wswz[(i * 2 + 0) * 64 + t];
            v2f a1 = s_wswz[(i * 2 + 1) * 64 + t];
            float w0 = half ? a0.y : a0.x;   // w[4t+2*half]
            float w1 = half ? a1.y : a1.x;   // w[4t+1+2*half]
            part += zc[t].x * w0 + zc[t].y * w1;
        }
        pre = part + __shfl_xor(part, 16, 32);
#endif
        float act = tanhf(pre + s_b[i]);
        const v2f* sc = scorr2 + i * (DIM / 2) + half;
#pragma unroll
        for (int t = 0; t < 64; ++t) {
            v2f sv = sc[t * 2];
            zc[t].x += sv.x * act;
            zc[t].y += sv.y * act;
        }
        float detp = 1.f + (1.f - act * act) * s_wsdot[i];
        loss += logf(fabsf(detp) + EPSF);

        // ======================= radial layer (2i+1) =====================
        const v2f* z0c = z02 + i * (DIM / 2) + half;
        float r2 = 0.f;
#pragma unroll
        for (int t = 0; t < 64; ++t) {
            v2f zv = zc[t], v0 = z0c[t * 2];
            float rx = zv.x - v0.x, ry = zv.y - v0.y;
            r2 += rx * rx + ry * ry;
        }
        r2 += __shfl_xor(r2, 16, 32);        // per-row ||radius||^2 (both halves)
        float r  = sqrtf(r2);
        float h  = 1.f / (s_a[i] + r);
        float bh = s_beta[i] * h;
#pragma unroll
        for (int t = 0; t < 64; ++t) {
            v2f v0 = z0c[t * 2];
            zc[t].x += bh * (zc[t].x - v0.x);
            zc[t].y += bh * (zc[t].y - v0.y);
        }
        if (half == 0) ws[OFF_H + (size_t)i * BATCH + grow] = h;   // defer R term
        // deterministic per-wave sum of the 16 row r^2 values
        float w2 = r2;
        w2 += __shfl_xor(w2, 1, 32);
        w2 += __shfl_xor(w2, 2, 32);
        w2 += __shfl_xor(w2, 4, 32);
        w2 += __shfl_xor(w2, 8, 32);
        if (lane == 0) s_wr2[wave][i] = w2;
    }

    v2f* zo2 = reinterpret_cast<v2f*>(zout) + (size_t)grow * (DIM / 2) + half;
#pragma unroll
    for (int t = 0; t < 64; ++t) zo2[t * 2] = zc[t];
    if (half == 0) ws[OFF_LOSSROW + grow] = loss;

    __syncthreads();
    if (tid < NPAIR) {   // fixed-order per-block r^2 partial (deterministic)
        float acc = 0.f;
        for (int wv = 0; wv < 8; ++wv) acc += s_wr2[wv][tid];
        ws[OFF_R2P + (size_t)blockIdx.x * NPAIR + tid] = acc;
    }
}

// ---- reduce r^2 partials -> R[q] = sqrt(sum) (1 block) ---------------------
__global__ __launch_bounds__(256) void flow_r2(float* __restrict__ ws) {
    __shared__ float red[256];
    const int t = threadIdx.x;
    for (int q = 0; q < NPAIR; ++q) {
        float acc = 0.f;
        for (int b = t; b < NBLK_MAIN; b += 256)
            acc += ws[OFF_R2P + (size_t)b * NPAIR + q];
        float s = bred(acc, red);
        if (t == 0) ws[OFF_R + q] = sqrtf(s);
        __syncthreads();
    }
}

// ---- per-row radial loss terms + per-block loss partial --------------------
__global__ __launch_bounds__(256) void flow_loss(float* __restrict__ ws) {
    __shared__ float red[256];
    const int row = blockIdx.x * 256 + threadIdx.x;
    float loss = ws[OFF_LOSSROW + row];
#pragma unroll
    for (int q = 0; q < NPAIR; ++q) {
        float h   = ws[OFF_H + (size_t)q * BATCH + row];
        float bh  = ws[OFF_BETA + q] * h;
        float Rq  = ws[OFF_R + q];
        float obh = 1.f + bh;                       // > 0 since beta >= -a
        float diag = expf(255.f * logf(obh));       // (1+bh)^(D-1)
        float det  = diag * (obh - bh * h * Rq);
        loss += logf(fabsf(det) + EPSF);
    }
    float s = bred(loss, red);
    if (threadIdx.x == 0) ws[OFF_LOSSPART + blockIdx.x] = s;
}

// ---- final mean ------------------------------------------------------------
__global__ __launch_bounds__(256) void flow_mean(const float* __restrict__ ws,
                                                 float* __restrict__ out) {
    __shared__ float red[256];
    float acc = 0.f;
    for (int b = threadIdx.x; b < NBLK_LOSS; b += 256)
        acc += ws[OFF_LOSSPART + b];
    float s = bred(acc, red);
    if (threadIdx.x == 0) out[(size_t)BATCH * DIM] = s / (float)BATCH;
}

extern "C" void kernel_launch(void* const* d_in, const int* in_sizes, int n_in,
                              void* d_out, int out_size, void* d_ws, size_t ws_size,
                              hipStream_t stream) {
    const float* z   = (const float*)d_in[0];
    const float* pw  = (const float*)d_in[1];
    const float* pb  = (const float*)d_in[2];
    const float* ps  = (const float*)d_in[3];
    const float* rz0 = (const float*)d_in[4];
    const float* ra  = (const float*)d_in[5];
    const float* rb  = (const float*)d_in[6];
    float* out = (float*)d_out;
    float* ws  = (float*)d_ws;

    flow_prep<<<1, 256, 0, stream>>>(pw, ps, ra, rb, ws);
    flow_main<<<NBLK_MAIN, 256, 0, stream>>>(z, pw, pb, rz0, ra, out, ws);
    flow_r2<<<1, 256, 0, stream>>>(ws);
    flow_loss<<<NBLK_LOSS, 256, 0, stream>>>(ws);
    flow_mean<<<1, 256, 0, stream>>>(ws, out);
}